// StackedEncoderModel_24945170055373
// MI455X (gfx1250) — compile-verified
//
#include <hip/hip_runtime.h>
#include <hip/hip_bf16.h>
#include <math.h>

// ---------------------------------------------------------------------------
// Problem constants (from reference)
// ---------------------------------------------------------------------------
#define LNUM 4
#define BB   4
#define TT   2048
#define DD   1024
#define HH   1024
#define FF   256
#define MM   (BB * TT)          // 8192 rows of activations

// scan chunking
#define NC   32                 // chunks over T
#define CL   (TT / NC)          // 64 steps per chunk

// GEMM tiling (double-buffered async Global->LDS)
#define BM   128
#define BN   128
#define BK   16
#define KPAD 20                 // 16 + 4 pad: rows are 80B (16B-aligned chunks), bank-safe

typedef float v2f __attribute__((ext_vector_type(2)));
typedef float v8f __attribute__((ext_vector_type(8)));

// ---------------------------------------------------------------------------
// Async Global->LDS copy path (gfx1250 GLOBAL_LOAD_ASYNC_TO_LDS_B128, ASYNCcnt)
// with a register-staged fallback if the builtin is unavailable.
// Builtin prototype (from hipcc diagnostic): param0 = v4i in addrspace(1),
// 4 args total.
// ---------------------------------------------------------------------------
#if __has_builtin(__builtin_amdgcn_global_load_async_to_lds_b128)
#define HAVE_ASYNC_LDS 1
#else
#define HAVE_ASYNC_LDS 0
#endif

typedef int v4i __attribute__((vector_size(16)));
typedef __attribute__((address_space(1))) v4i global_v4i;
typedef __attribute__((address_space(3))) v4i lds_v4i;

__device__ __forceinline__ void cp16_to_lds(const float* g, float* l)
{
#if HAVE_ASYNC_LDS
    __builtin_amdgcn_global_load_async_to_lds_b128(
        (global_v4i*)g, (lds_v4i*)l, 0, 0);
#else
    *(float4*)l = *(const float4*)g;
#endif
}

template <int N>
__device__ __forceinline__ void async_wait()
{
#if HAVE_ASYNC_LDS
#if __has_builtin(__builtin_amdgcn_s_wait_asynccnt)
    __builtin_amdgcn_s_wait_asynccnt((unsigned short)N);
#else
    asm volatile("s_wait_asynccnt %0" :: "i"(N));
#endif
#endif
}

// ---------------------------------------------------------------------------
// Generic NT GEMM:  Cout[m,n] (+)= alpha * nscale[n] * sum_k A[m,k]*Bw[n,k] + bias[n]
// A: [M,K] row-major, Bw: [N,K] row-major (torch Linear layout).
// M % 128 == 0, N % 128 == 0, K % 16 == 0 (true for all calls here).
// V_WMMA_F32_16X16X4_F32 compute, double-buffered async Global->LDS staging.
// ---------------------------------------------------------------------------
__global__ __launch_bounds__(256)
void gemm_nt_wmma(const float* __restrict__ A, const float* __restrict__ Bw,
                  float* __restrict__ Cout, int M, int N, int K,
                  float alpha, const float* __restrict__ bias,
                  const float* __restrict__ nscale, int accumulate)
{
    __shared__ float sA[2][BM * KPAD];   // 2 x 10 KB
    __shared__ float sB[2][BN * KPAD];   // 2 x 10 KB   (40 KB total)

    const int n0 = blockIdx.x * BN;
    const int m0 = blockIdx.y * BM;

    const int tid    = threadIdx.x;
    const int wave   = tid >> 5;        // 0..7  (wave32)
    const int lane   = tid & 31;
    const int wm     = wave >> 1;       // 0..3  -> 32-row slab
    const int wn     = wave & 1;        // 0..1  -> 64-col slab
    const int lane15 = lane & 15;
    const int laneHi = lane >> 4;

    // this thread's two 16B chunks per tile (A) and two per tile (B)
    const int slot0 = tid;              // 0..255
    const int slot1 = tid + 256;        // 256..511
    const int rowA0 = slot0 >> 2, cA0 = (slot0 & 3) * 4;
    const int rowA1 = slot1 >> 2, cA1 = (slot1 & 3) * 4;

    v8f acc[2][4];
#pragma unroll
    for (int mi = 0; mi < 2; ++mi)
#pragma unroll
        for (int ni = 0; ni < 4; ++ni)
            acc[mi][ni] = (v8f)0.0f;

    const int nkt = K / BK;

    // prologue: stage tile 0 into buffer 0
    {
        const int kb = 0;
        cp16_to_lds(A  + (size_t)(m0 + rowA0) * K + kb + cA0, &sA[0][rowA0 * KPAD + cA0]);
        cp16_to_lds(A  + (size_t)(m0 + rowA1) * K + kb + cA1, &sA[0][rowA1 * KPAD + cA1]);
        cp16_to_lds(Bw + (size_t)(n0 + rowA0) * K + kb + cA0, &sB[0][rowA0 * KPAD + cA0]);
        cp16_to_lds(Bw + (size_t)(n0 + rowA1) * K + kb + cA1, &sB[0][rowA1 * KPAD + cA1]);
    }

    for (int kt = 0; kt < nkt; ++kt) {
        const int cur = kt & 1;
        const int nxt = cur ^ 1;
        const bool has_next = (kt + 1) < nkt;

        // issue async copy of next tile into the other buffer (overlaps compute)
        if (has_next) {
            const int kb = (kt + 1) * BK;
            cp16_to_lds(A  + (size_t)(m0 + rowA0) * K + kb + cA0, &sA[nxt][rowA0 * KPAD + cA0]);
            cp16_to_lds(A  + (size_t)(m0 + rowA1) * K + kb + cA1, &sA[nxt][rowA1 * KPAD + cA1]);
            cp16_to_lds(Bw + (size_t)(n0 + rowA0) * K + kb + cA0, &sB[nxt][rowA0 * KPAD + cA0]);
            cp16_to_lds(Bw + (size_t)(n0 + rowA1) * K + kb + cA1, &sB[nxt][rowA1 * KPAD + cA1]);
            async_wait<4>();            // current tile's 4 copies retired (in-order)
        } else {
            async_wait<0>();            // all copies retired
        }
        __syncthreads();                // all waves' copies to buf[cur] visible

        // --- 4 k-steps of 4, 2x4 WMMA tiles per wave
#pragma unroll
        for (int k0 = 0; k0 < BK; k0 += 4) {
            const int kk = k0 + laneHi * 2;     // fragment element v holds k = kk+v
            v2f afrag[2], bfrag[4];
#pragma unroll
            for (int mi = 0; mi < 2; ++mi) {
                const int r = wm * 32 + mi * 16 + lane15;
                afrag[mi] = *(const v2f*)(&sA[cur][r * KPAD + kk]);
            }
#pragma unroll
            for (int ni = 0; ni < 4; ++ni) {
                const int c = wn * 64 + ni * 16 + lane15;
                bfrag[ni] = *(const v2f*)(&sB[cur][c * KPAD + kk]);
            }
#pragma unroll
            for (int mi = 0; mi < 2; ++mi)
#pragma unroll
                for (int ni = 0; ni < 4; ++ni)
                    acc[mi][ni] = __builtin_amdgcn_wmma_f32_16x16x4_f32(
                        false, afrag[mi], false, bfrag[ni],
                        (short)0, acc[mi][ni], false, false);
        }
        __syncthreads();                // safe to overwrite buf[cur] next round
    }

    // --- epilogue (C/D layout: VGPR r -> m_local = r + 8*(lane>>4), n = lane&15)
#pragma unroll
    for (int mi = 0; mi < 2; ++mi) {
#pragma unroll
        for (int ni = 0; ni < 4; ++ni) {
            const int n  = n0 + wn * 64 + ni * 16 + lane15;
            const float sc = alpha * (nscale ? nscale[n] : 1.0f);
            const float bz = bias ? bias[n] : 0.0f;
#pragma unroll
            for (int r = 0; r < 8; ++r) {
                const int m = m0 + wm * 32 + mi * 16 + laneHi * 8 + r;
                const size_t o = (size_t)m * N + n;
                float v = acc[mi][ni][r] * sc + bz;
                if (accumulate) v += Cout[o];
                Cout[o] = v;
            }
        }
    }
}

// ---------------------------------------------------------------------------
// LayerNorm over last dim (D=1024), one 256-thread block per row.
// ---------------------------------------------------------------------------
__global__ __launch_bounds__(256)
void ln_kernel(const float* __restrict__ x, float* __restrict__ xn,
               const float* __restrict__ scale, const float* __restrict__ bias)
{
    __shared__ float red[256];
    const int row = blockIdx.x;
    const int tid = threadIdx.x;
    const size_t base = (size_t)row * DD;

    float s = 0.f, s2 = 0.f;
    float v[4];
#pragma unroll
    for (int i = 0; i < 4; ++i) {
        v[i] = x[base + tid + i * 256];
        s  += v[i];
        s2 += v[i] * v[i];
    }
    red[tid] = s;
    __syncthreads();
    for (int off = 128; off > 0; off >>= 1) {
        if (tid < off) red[tid] += red[tid + off];
        __syncthreads();
    }
    const float mu = red[0] * (1.0f / DD);
    __syncthreads();
    red[tid] = s2;
    __syncthreads();
    for (int off = 128; off > 0; off >>= 1) {
        if (tid < off) red[tid] += red[tid + off];
        __syncthreads();
    }
    const float var = red[0] * (1.0f / DD) - mu * mu;
    const float inv = rsqrtf(var + 1e-5f);
#pragma unroll
    for (int i = 0; i < 4; ++i) {
        const int d = tid + i * 256;
        xn[base + d] = (v[i] - mu) * inv * scale[d] + bias[d];
    }
}

// ---------------------------------------------------------------------------
// Per-layer LRU parameter prep: lambda (complex) and gamma, H threads.
// ---------------------------------------------------------------------------
__global__ void prep_lru(const float* __restrict__ nu_log,
                         const float* __restrict__ theta_log,
                         float* __restrict__ lam_re, float* __restrict__ lam_im,
                         float* __restrict__ gamma)
{
    const int h = blockIdx.x * blockDim.x + threadIdx.x;
    if (h >= HH) return;
    const float nu = expf(nu_log[h]);
    const float th = expf(theta_log[h]);
    const float r  = expf(-nu);
    lam_re[h] = r * cosf(th);
    lam_im[h] = r * sinf(th);
    gamma[h]  = sqrtf(1.0f - expf(-2.0f * nu) + 1e-5f);
}

// ---------------------------------------------------------------------------
// Chunked parallel scan: h_t = lam * h_{t-1} + bu_t  (complex, in place).
// ---------------------------------------------------------------------------
__global__ __launch_bounds__(256)
void scan1(float* __restrict__ hre, float* __restrict__ him,
           const float* __restrict__ lam_re, const float* __restrict__ lam_im,
           float* __restrict__ lastre, float* __restrict__ lastim)
{
    const int idx = blockIdx.x * blockDim.x + threadIdx.x;   // B*NC*H threads
    const int h = idx & (HH - 1);
    const int c = (idx >> 10) & (NC - 1);
    const int b = idx >> 15;
    const float lr = lam_re[h], li = lam_im[h];
    const size_t base = ((size_t)(b * TT + c * CL)) * HH + h;
    float ar = 0.f, ai = 0.f;
    for (int t = 0; t < CL; ++t) {
        const size_t p = base + (size_t)t * HH;
        const float br = hre[p], bi = him[p];
        const float nr = lr * ar - li * ai + br;
        const float ni = lr * ai + li * ar + bi;
        hre[p] = nr; him[p] = ni;
        ar = nr; ai = ni;
    }
    lastre[(b * NC + c) * HH + h] = ar;
    lastim[(b * NC + c) * HH + h] = ai;
}

__global__ __launch_bounds__(256)
void scan2(const float* __restrict__ nu_log, const float* __restrict__ theta_log,
           const float* __restrict__ lastre, const float* __restrict__ lastim,
           float* __restrict__ prefre, float* __restrict__ prefim)
{
    const int idx = blockIdx.x * blockDim.x + threadIdx.x;   // B*H threads
    const int h = idx & (HH - 1);
    const int b = idx >> 10;
    const float nu = expf(nu_log[h]);
    const float th = expf(theta_log[h]);
    const float pr = expf(-(float)CL * nu);          // |lam|^CL
    const float an = (float)CL * th;                 // CL * theta
    const float Pr = pr * cosf(an), Pi = pr * sinf(an);
    float cr = 0.f, ci = 0.f;
    for (int c = 0; c < NC; ++c) {
        const int o = (b * NC + c) * HH + h;
        prefre[o] = cr; prefim[o] = ci;
        const float lr2 = lastre[o], li2 = lastim[o];
        const float nr = Pr * cr - Pi * ci + lr2;
        const float ni = Pr * ci + Pi * cr + li2;
        cr = nr; ci = ni;
    }
}

__global__ __launch_bounds__(256)
void scan3(float* __restrict__ hre, float* __restrict__ him,
           const float* __restrict__ lam_re, const float* __restrict__ lam_im,
           const float* __restrict__ prefre, const float* __restrict__ prefim)
{
    const int idx = blockIdx.x * blockDim.x + threadIdx.x;   // B*NC*H threads
    const int h = idx & (HH - 1);
    const int c = (idx >> 10) & (NC - 1);
    const int b = idx >> 15;
    const float cr = prefre[(b * NC + c) * HH + h];
    const float ci = prefim[(b * NC + c) * HH + h];
    if (cr == 0.f && ci == 0.f) return;              // chunk 0 & fully decayed carries
    const float lr = lam_re[h], li = lam_im[h];
    float mr = lr * cr - li * ci;                    // lam^1 * carry
    float mi = lr * ci + li * cr;
    const size_t base = ((size_t)(b * TT + c * CL)) * HH + h;
    for (int t = 0; t < CL; ++t) {
        const size_t p = base + (size_t)t * HH;
        hre[p] += mr; him[p] += mi;
        const float nr = lr * mr - li * mi;
        const float ni = lr * mi + li * mr;
        mr = nr; mi = ni;
    }
}

// ---------------------------------------------------------------------------
// y = gelu_exact(y + xn * D_diag[d])
// ---------------------------------------------------------------------------
__global__ __launch_bounds__(256)
void ddiag_gelu(float* __restrict__ y, const float* __restrict__ xn,
                const float* __restrict__ D_diag)
{
    const size_t idx = (size_t)blockIdx.x * blockDim.x + threadIdx.x;
    const int d = (int)(idx & (DD - 1));
    const float v = y[idx] + xn[idx] * D_diag[d];
    y[idx] = 0.5f * v * (1.0f + erff(v * 0.70710678118654752f));
}

// ---------------------------------------------------------------------------
// x += g1 * sigmoid(g2)   (GLU + residual)
// ---------------------------------------------------------------------------
__global__ __launch_bounds__(256)
void glu_residual(float* __restrict__ x, const float* __restrict__ g1,
                  const float* __restrict__ g2)
{
    const size_t idx = (size_t)blockIdx.x * blockDim.x + threadIdx.x;
    const float a = g1[idx];
    const float s = 1.0f / (1.0f + expf(-g2[idx]));
    x[idx] += a * s;
}

// ---------------------------------------------------------------------------
// Launcher
// ---------------------------------------------------------------------------
extern "C" void kernel_launch(void* const* d_in, const int* in_sizes, int n_in,
                              void* d_out, int out_size, void* d_ws, size_t ws_size,
                              hipStream_t stream)
{
    (void)in_sizes; (void)n_in; (void)out_size; (void)ws_size;
    const float* inputs    = (const float*)d_in[0];
    const float* W_in      = (const float*)d_in[1];
    const float* b_in      = (const float*)d_in[2];
    const float* nu_log    = (const float*)d_in[3];
    const float* theta_log = (const float*)d_in[4];
    const float* B_re      = (const float*)d_in[5];
    const float* B_im      = (const float*)d_in[6];
    const float* C_re      = (const float*)d_in[7];
    const float* C_im      = (const float*)d_in[8];
    const float* D_diag    = (const float*)d_in[9];
    const float* ln_scale  = (const float*)d_in[10];
    const float* ln_bias   = (const float*)d_in[11];
    const float* w1        = (const float*)d_in[12];
    const float* b1        = (const float*)d_in[13];
    const float* w2        = (const float*)d_in[14];
    const float* b2        = (const float*)d_in[15];

    const size_t SZ = (size_t)MM * DD;               // 8M floats per activation
    float* ws = (float*)d_ws;
    float* x     = ws;                               // persistent residual stream
    float* xn    = ws + 1 * SZ;
    float* bu_re = ws + 2 * SZ;                      // becomes h_re; reused as g1
    float* bu_im = ws + 3 * SZ;                      // becomes h_im; reused as g2
    float* y     = ws + 4 * SZ;
    float* small = ws + 5 * SZ;
    float* lam_re = small;
    float* lam_im = small + HH;
    float* gamvec = small + 2 * HH;
    const size_t BNCH = (size_t)BB * NC * HH;        // 131072
    float* last_re = small + 3 * HH;
    float* last_im = last_re + BNCH;
    float* pref_re = last_im + BNCH;
    float* pref_im = pref_re + BNCH;

    const dim3 blk(256);
    const dim3 gemm_grid(DD / BN, MM / BM);          // (8, 64)
    const dim3 ew_grid((unsigned)(SZ / 256));        // elementwise over B*T*D
    const dim3 scan_grid((unsigned)(BNCH / 256));    // B*NC*H threads

    // x = inputs @ W_in^T + b_in   (M=8192, N=1024, K=256)
    gemm_nt_wmma<<<gemm_grid, blk, 0, stream>>>(inputs, W_in, x, MM, DD, FF,
                                                1.0f, b_in, nullptr, 0);

    for (int l = 0; l < LNUM; ++l) {
        const float* nu_l  = nu_log    + (size_t)l * HH;
        const float* th_l  = theta_log + (size_t)l * HH;
        const float* Bre_l = B_re + (size_t)l * HH * DD;
        const float* Bim_l = B_im + (size_t)l * HH * DD;
        const float* Cre_l = C_re + (size_t)l * DD * HH;
        const float* Cim_l = C_im + (size_t)l * DD * HH;
        const float* Dd_l  = D_diag   + (size_t)l * DD;
        const float* lns_l = ln_scale + (size_t)l * DD;
        const float* lnb_l = ln_bias  + (size_t)l * DD;
        const float* w1_l  = w1 + (size_t)l * DD * DD;
        const float* w2_l  = w2 + (size_t)l * DD * DD;
        const float* b1_l  = b1 + (size_t)l * DD;
        const float* b2_l  = b2 + (size_t)l * DD;

        // LRU scalars for this layer
        prep_lru<<<dim3(HH / 256), blk, 0, stream>>>(nu_l, th_l, lam_re, lam_im, gamvec);

        // xn = LayerNorm(x)
        ln_kernel<<<dim3(MM), blk, 0, stream>>>(x, xn, lns_l, lnb_l);

        // Bu = gamma[h] * (xn @ B_{re,im}^T)    (K = D = 1024)
        gemm_nt_wmma<<<gemm_grid, blk, 0, stream>>>(xn, Bre_l, bu_re, MM, HH, DD,
                                                    1.0f, nullptr, gamvec, 0);
        gemm_nt_wmma<<<gemm_grid, blk, 0, stream>>>(xn, Bim_l, bu_im, MM, HH, DD,
                                                    1.0f, nullptr, gamvec, 0);

        // complex scan over T (in place: bu -> h)
        scan1<<<scan_grid, blk, 0, stream>>>(bu_re, bu_im, lam_re, lam_im, last_re, last_im);
        scan2<<<dim3(BB * HH / 256), blk, 0, stream>>>(nu_l, th_l, last_re, last_im,
                                                       pref_re, pref_im);
        scan3<<<scan_grid, blk, 0, stream>>>(bu_re, bu_im, lam_re, lam_im, pref_re, pref_im);

        // y = Re(h @ C^T) = h_re @ C_re^T - h_im @ C_im^T   (K = H = 1024)
        gemm_nt_wmma<<<gemm_grid, blk, 0, stream>>>(bu_re, Cre_l, y, MM, DD, HH,
                                                    1.0f, nullptr, nullptr, 0);
        gemm_nt_wmma<<<gemm_grid, blk, 0, stream>>>(bu_im, Cim_l, y, MM, DD, HH,
                                                    -1.0f, nullptr, nullptr, 1);

        // y = gelu(y + xn * D_diag)
        ddiag_gelu<<<ew_grid, blk, 0, stream>>>(y, xn, Dd_l);

        // GLU: g1 = y@w1^T+b1, g2 = y@w2^T+b2 (reuse bu buffers), x += g1*sigmoid(g2)
        gemm_nt_wmma<<<gemm_grid, blk, 0, stream>>>(y, w1_l, bu_re, MM, DD, DD,
                                                    1.0f, b1_l, nullptr, 0);
        gemm_nt_wmma<<<gemm_grid, blk, 0, stream>>>(y, w2_l, bu_im, MM, DD, DD,
                                                    1.0f, b2_l, nullptr, 0);
        glu_residual<<<ew_grid, blk, 0, stream>>>(x, bu_re, bu_im);
    }

    (void)hipMemcpyAsync(d_out, x, SZ * sizeof(float), hipMemcpyDeviceToDevice, stream);
}